// GVF_MoE_Final_Model_21543555957371
// MI455X (gfx1250) — compile-verified
//
#include <hip/hip_runtime.h>
#include <math.h>

// ---------------------------------------------------------------------------
// GVF MoE (GAT env/behav experts + EvolveGCN-O phys expert + gating) for
// MI455X / gfx1250.  Node GEMMs run fully on the WMMA pipe in fp32
// (v_wmma_f32_16x16x4_f32, one wave32 per 16x16 tile, compile-time K so the
// whole K loop unrolls into back-to-back WMMAs with unpredicated b64 loads).
// Edge phase uses hardware float atomics (global_atomic_add_f32) and an
// order-preserving uint encoding for segment-max (global_atomic_max_u32).
// ---------------------------------------------------------------------------

typedef __attribute__((ext_vector_type(2))) float v2f;
typedef __attribute__((ext_vector_type(8))) float v8f;

#define NNODES 100000            // multiple of 16 -> no row guards in GEMM
#define NEDGES 1600000
#define ROWTILES (NNODES / 16)   // 6250
#define WAVES_PER_BLK 8

// ---- order-preserving float <-> uint encoding (for atomic segment-max) ----
__device__ __forceinline__ unsigned enc_f(float x) {
  unsigned b = __float_as_uint(x);
  return (b & 0x80000000u) ? ~b : (b | 0x80000000u);
}
__device__ __forceinline__ float dec_f(unsigned u) {
  return (u & 0x80000000u) ? __uint_as_float(u & 0x7FFFFFFFu)
                           : __uint_as_float(~u);
}

// ---------------------------------------------------------------------------
// fp32 WMMA GEMM:  C[N, Nout] = act(A[N, K] @ Bp[K, 16*gridDim.y] + biasp)
//  * K is a compile-time multiple of 4; A rows are zero-padded to lda (>=K,
//    multiple of 4) so loads need no guards.
//  * Bp / biasp are zero-padded staging buffers of width 16*colTiles.
//  * One wave per 16x16 tile, 8 waves per block.
//  * act: 0 = none, 1 = relu, 2 = elu.
// A lane layout (16x4 f32): lane%16 = row, lane/16 selects K pair (VGPR0/1).
// B lane layout (4x16 f32): lane%16 = col, lane/16 selects K pair.
// C/D (16x16 f32): lane%16 = col N, row M = 8*(lane/16) + vgpr.
// ---------------------------------------------------------------------------
template <int K>
__global__ void gemm_wmma_f32(const float* __restrict__ A, int lda,
                              const float* __restrict__ Bp,
                              const float* __restrict__ biasp,
                              float* __restrict__ C, int ldc,
                              int Nout, int act) {
  const int lane = threadIdx.x & 31;
  const int wv = threadIdx.x >> 5;
  const int tile = blockIdx.x * WAVES_PER_BLK + wv;
  if (tile >= ROWTILES) return;                 // whole-wave uniform exit
  const int half = lane >> 4;                   // 0 or 1
  const int l16 = lane & 15;
  const int r0 = tile * 16;
  const int bcol = blockIdx.y * 16 + l16;
  const int npad = (int)gridDim.y * 16;         // Bp leading dimension
  const float* __restrict__ Arow = A + (size_t)(r0 + l16) * lda;

  v8f acc = {0.f, 0.f, 0.f, 0.f, 0.f, 0.f, 0.f, 0.f};
#pragma unroll
  for (int k0 = 0; k0 < K; k0 += 4) {
    const int ka = k0 + half * 2;
    v2f a = *(const v2f*)(Arow + ka);           // one global_load_b64
    v2f b;
    b.x = Bp[(size_t)ka * npad + bcol];
    b.y = Bp[(size_t)(ka + 1) * npad + bcol];
    acc = __builtin_amdgcn_wmma_f32_16x16x4_f32(
        false, a, false, b, (short)0, acc, false, false);
  }

  if (bcol >= Nout) return;                     // store guard only
  const float bv = biasp[bcol];
#pragma unroll
  for (int v = 0; v < 8; ++v) {
    int row = r0 + half * 8 + v;
    float x = acc[v] + bv;
    if (act == 1) x = x > 0.f ? x : 0.f;
    else if (act == 2) x = x > 0.f ? x : expm1f(x);
    C[(size_t)row * ldc + bcol] = x;
  }
}

// ---- zero-padded weight/bias staging: Bp[Kpad,Npad], biasp[Npad] ----
__global__ void pad_wb(const float* __restrict__ B, const float* __restrict__ bias,
                       int K, int Nout, float* __restrict__ Bp,
                       float* __restrict__ biasp, int Kpad, int Npad) {
  int tid = blockIdx.x * blockDim.x + threadIdx.x;
  if (tid < Kpad * Npad) {
    int k = tid / Npad, n = tid % Npad;
    Bp[tid] = (k < K && n < Nout) ? B[k * Nout + n] : 0.f;
  }
  if (tid < Npad) biasp[tid] = (bias != nullptr && tid < Nout) ? bias[tid] : 0.f;
}

// ---- pad raw x [N,6] -> [N,8] (zero cols 6,7) ----
__global__ void pad_x(const float* __restrict__ x, float* __restrict__ xp8) {
  int i = blockIdx.x * blockDim.x + threadIdx.x;
  if (i >= NNODES) return;
  const float* xi = x + (size_t)i * 6;
  float* o = xp8 + (size_t)i * 8;
  o[0] = xi[0]; o[1] = xi[1]; o[2] = xi[2];
  o[3] = xi[3]; o[4] = xi[4]; o[5] = xi[5];
  o[6] = 0.f;   o[7] = 0.f;
}

// ---- per-node attention logits: a_src/a_dst[i,h] = <h[i,h,:], att[h,:]> ----
__global__ void att_scores(const float* __restrict__ h,
                           const float* __restrict__ attS,
                           const float* __restrict__ attD,
                           float* __restrict__ aS, float* __restrict__ aD,
                           int n, int H, int C) {
  int tid = blockIdx.x * blockDim.x + threadIdx.x;
  if (tid >= n * H) return;
  int i = tid / H, hd = tid % H;
  const float* hp = h + (size_t)i * H * C + hd * C;
  const float* ws = attS + hd * C;
  const float* wd = attD + hd * C;
  float s = 0.f, d = 0.f;
  for (int c = 0; c < C; ++c) { float v = hp[c]; s += v * ws[c]; d += v * wd[c]; }
  aS[tid] = s; aD[tid] = d;
}

__global__ void init_u32(unsigned* __restrict__ p, unsigned v, int n) {
  int tid = blockIdx.x * blockDim.x + threadIdx.x;
  if (tid < n) p[tid] = v;
}

// ---- GAT edge pass 1: e = leakyrelu(a_src[s]+a_dst[d]); segment max ----
__global__ void gat_edge_pass1(const int* __restrict__ ei,
                               const float* __restrict__ aS,
                               const float* __restrict__ aD,
                               float* __restrict__ ev, unsigned* __restrict__ m,
                               int H) {
  int e = blockIdx.x * blockDim.x + threadIdx.x;
  if (e >= NEDGES) return;
  int s = ei[e], d = ei[NEDGES + e];
  for (int h = 0; h < H; ++h) {
    float v = aS[s * H + h] + aD[d * H + h];
    v = v > 0.f ? v : 0.2f * v;
    ev[(size_t)e * H + h] = v;
    atomicMax(&m[d * H + h], enc_f(v));
  }
}
__global__ void gat_self_pass1(const float* __restrict__ aS,
                               const float* __restrict__ aD,
                               float* __restrict__ evs, unsigned* __restrict__ m,
                               int H) {
  int i = blockIdx.x * blockDim.x + threadIdx.x;
  if (i >= NNODES) return;
  for (int h = 0; h < H; ++h) {
    float v = aS[i * H + h] + aD[i * H + h];
    v = v > 0.f ? v : 0.2f * v;
    evs[(size_t)i * H + h] = v;
    atomicMax(&m[i * H + h], enc_f(v));
  }
}

// ---- GAT edge pass 2: e = exp(e - m[d]); segment sum -> z ----
__global__ void gat_edge_pass2(const int* __restrict__ ei,
                               float* __restrict__ ev,
                               const unsigned* __restrict__ m,
                               float* __restrict__ z, int H) {
  int e = blockIdx.x * blockDim.x + threadIdx.x;
  if (e >= NEDGES) return;
  int d = ei[NEDGES + e];
  for (int h = 0; h < H; ++h) {
    float ex = __expf(ev[(size_t)e * H + h] - dec_f(m[d * H + h]));
    ev[(size_t)e * H + h] = ex;
    atomicAdd(&z[d * H + h], ex);
  }
}
__global__ void gat_self_pass2(float* __restrict__ evs,
                               const unsigned* __restrict__ m,
                               float* __restrict__ z, int H) {
  int i = blockIdx.x * blockDim.x + threadIdx.x;
  if (i >= NNODES) return;
  for (int h = 0; h < H; ++h) {
    float ex = __expf(evs[(size_t)i * H + h] - dec_f(m[i * H + h]));
    evs[(size_t)i * H + h] = ex;
    atomicAdd(&z[i * H + h], ex);
  }
}

// ---- GAT edge pass 3: out[d] += (e/z[d]) * h[s] ----
__global__ void gat_edge_pass3(const int* __restrict__ ei,
                               const float* __restrict__ ev,
                               const float* __restrict__ z,
                               const float* __restrict__ hf,
                               float* __restrict__ out, int H, int C) {
  int e = blockIdx.x * blockDim.x + threadIdx.x;
  if (e >= NEDGES) return;
  int s = ei[e], d = ei[NEDGES + e];
  for (int h = 0; h < H; ++h) {
    float alpha = ev[(size_t)e * H + h] / z[d * H + h];
    const float* hp = hf + ((size_t)s * H + h) * C;
    float* op = out + ((size_t)d * H + h) * C;
    for (int c = 0; c < C; c += 4) {
      float4 hv = *(const float4*)(hp + c);
      atomicAdd(op + c + 0, alpha * hv.x);
      atomicAdd(op + c + 1, alpha * hv.y);
      atomicAdd(op + c + 2, alpha * hv.z);
      atomicAdd(op + c + 3, alpha * hv.w);
    }
  }
}
__global__ void gat_self_pass3(const float* __restrict__ evs,
                               const float* __restrict__ z,
                               const float* __restrict__ hf,
                               float* __restrict__ out, int H, int C) {
  int i = blockIdx.x * blockDim.x + threadIdx.x;
  if (i >= NNODES) return;
  for (int h = 0; h < H; ++h) {
    float alpha = evs[(size_t)i * H + h] / z[i * H + h];
    const float* hp = hf + ((size_t)i * H + h) * C;
    float* op = out + ((size_t)i * H + h) * C;
    for (int c = 0; c < C; c += 4) {
      float4 hv = *(const float4*)(hp + c);
      atomicAdd(op + c + 0, alpha * hv.x);
      atomicAdd(op + c + 1, alpha * hv.y);
      atomicAdd(op + c + 2, alpha * hv.z);
      atomicAdd(op + c + 3, alpha * hv.w);
    }
  }
}

// ---- bias add (+ optional elu) on aggregated conv output ----
__global__ void bias_act(float* __restrict__ x, const float* __restrict__ bias,
                         int n, int F, int act) {
  size_t tid = (size_t)blockIdx.x * blockDim.x + threadIdx.x;
  if (tid >= (size_t)n * F) return;
  int j = (int)(tid % F);
  float v = x[tid] + bias[j];
  if (act == 2) v = v > 0.f ? v : expm1f(v);
  x[tid] = v;
}

__global__ void relu_inplace(float* __restrict__ x, size_t n) {
  size_t tid = (size_t)blockIdx.x * blockDim.x + threadIdx.x;
  if (tid >= n) return;
  float v = x[tid];
  x[tid] = v > 0.f ? v : 0.f;
}

// ---- build x_env/x_behav [N, stride 12] and x_phys [N, stride 12] ----
__global__ void prep_inputs(const float* __restrict__ x,
                            const float* __restrict__ ctx,
                            float* __restrict__ xe, float* __restrict__ xp,
                            float* __restrict__ xb) {
  int i = blockIdx.x * blockDim.x + threadIdx.x;
  if (i >= NNODES) return;
  const float* xi = x + (size_t)i * 6;
  float* e = xe + (size_t)i * 12;
  float* p = xp + (size_t)i * 12;
  float* b = xb + (size_t)i * 12;
  e[0] = xi[0]; e[1] = xi[1]; e[2] = xi[5];
  b[0] = xi[0]; b[1] = xi[1]; b[2] = xi[2];
  p[0] = xi[3]; p[1] = xi[4];
  for (int k = 0; k < 8; ++k) {
    float cv = ctx[(size_t)i * 8 + k];
    e[3 + k] = cv; b[3 + k] = cv; p[2 + k] = cv;
  }
  e[11] = 0.f; b[11] = 0.f; p[10] = 0.f; p[11] = 0.f;
}

// ---- EvolveGCN-O weight RNN; writes zero-padded B^T [12 x 16] ----
// BTp[k*16 + n] = W_new[n][k], with W_new = tanh(W0_flat @ Wih^T + bih + bhh).
__global__ void phys_rnn(const float* __restrict__ W0,
                         const float* __restrict__ Wih,
                         const float* __restrict__ bih,
                         const float* __restrict__ bhh,
                         float* __restrict__ BTp) {
  int j = threadIdx.x;
  if (j < 12 * 16) BTp[j] = 0.f;
  __syncthreads();
  if (j >= 100) return;
  float acc = bih[j] + bhh[j];
  for (int k = 0; k < 100; ++k) acc += W0[k] * Wih[j * 100 + k];
  float h = tanhf(acc);          // h == W_new[j/10][j%10]
  int r = j / 10, c = j % 10;    // r = output row n, c = contraction k
  BTp[c * 16 + r] = h;
}

// ---- GCN propagate: agg[d] += xp[s] (no self loops) ----
__global__ void gcn_edge(const int* __restrict__ ei, const float* __restrict__ xp,
                         float* __restrict__ agg) {
  int e = blockIdx.x * blockDim.x + threadIdx.x;
  if (e >= NEDGES) return;
  int s = ei[e], d = ei[NEDGES + e];
  const float* sp = xp + (size_t)s * 10;
  float* dp = agg + (size_t)d * 12;          // agg rows padded to 12
  for (int c = 0; c < 10; ++c) atomicAdd(dp + c, sp[c]);
}

// ---- gate softmax (3-way) + expert mixture -> final output [N,16] ----
__global__ void gate_combine(const float* __restrict__ gsc,
                             const float* __restrict__ env,
                             const float* __restrict__ phys,
                             const float* __restrict__ behav,
                             float* __restrict__ out) {
  int i = blockIdx.x * blockDim.x + threadIdx.x;
  if (i >= NNODES) return;
  float a = gsc[i * 3 + 0], b = gsc[i * 3 + 1], c = gsc[i * 3 + 2];
  float mx = fmaxf(a, fmaxf(b, c));
  float ea = __expf(a - mx), eb = __expf(b - mx), ec = __expf(c - mx);
  float inv = 1.f / (ea + eb + ec);
  float w0 = ea * inv, w1 = eb * inv, w2 = ec * inv;
  size_t base = (size_t)i * 16;
  for (int k = 0; k < 16; ++k)
    out[base + k] = w0 * env[base + k] + w1 * phys[base + k] + w2 * behav[base + k];
}

// ---------------------------------------------------------------------------
// Host-side orchestration
// ---------------------------------------------------------------------------
static inline dim3 gemm_grid(int colTiles) {
  return dim3((ROWTILES + WAVES_PER_BLK - 1) / WAVES_PER_BLK, colTiles);
}

static void run_gat_layer(hipStream_t stream, const int* ei,
                          const float* xin, int lda, int kpad, int H, int C,
                          const float* lin, int Kin,
                          const float* attS, const float* attD,
                          const float* bias,
                          float* Bp, float* biasp,
                          float* hbuf, float* aS, float* aD, unsigned* m,
                          float* z, float* evE, float* evS, float* outAcc,
                          int act) {
  const int HC = H * C;
  const int colTiles = HC / 16;
  pad_wb<<<4, 256, 0, stream>>>(lin, nullptr, Kin, HC, Bp, biasp, kpad, HC);
  if (kpad == 12)
    gemm_wmma_f32<12><<<gemm_grid(colTiles), 32 * WAVES_PER_BLK, 0, stream>>>(
        xin, lda, Bp, biasp, hbuf, HC, HC, 0);
  else
    gemm_wmma_f32<32><<<gemm_grid(colTiles), 32 * WAVES_PER_BLK, 0, stream>>>(
        xin, lda, Bp, biasp, hbuf, HC, HC, 0);
  const int nh = NNODES * H;
  att_scores<<<(nh + 255) / 256, 256, 0, stream>>>(hbuf, attS, attD, aS, aD,
                                                   NNODES, H, C);
  init_u32<<<(nh + 255) / 256, 256, 0, stream>>>(m, 0x007FFFFFu, nh); // enc(-inf)
  hipMemsetAsync(z, 0, (size_t)nh * sizeof(float), stream);
  hipMemsetAsync(outAcc, 0, (size_t)NNODES * HC * sizeof(float), stream);
  const int eb = (NEDGES + 255) / 256, nb = (NNODES + 255) / 256;
  gat_edge_pass1<<<eb, 256, 0, stream>>>(ei, aS, aD, evE, m, H);
  gat_self_pass1<<<nb, 256, 0, stream>>>(aS, aD, evS, m, H);
  gat_edge_pass2<<<eb, 256, 0, stream>>>(ei, evE, m, z, H);
  gat_self_pass2<<<nb, 256, 0, stream>>>(evS, m, z, H);
  gat_edge_pass3<<<eb, 256, 0, stream>>>(ei, evE, z, hbuf, outAcc, H, C);
  gat_self_pass3<<<nb, 256, 0, stream>>>(evS, z, hbuf, outAcc, H, C);
  bias_act<<<(unsigned)(((size_t)NNODES * HC + 255) / 256), 256, 0, stream>>>(
      outAcc, bias, NNODES, HC, act);
}

extern "C" void kernel_launch(void* const* d_in, const int* in_sizes, int n_in,
                              void* d_out, int out_size, void* d_ws,
                              size_t ws_size, hipStream_t stream) {
  (void)in_sizes; (void)n_in; (void)out_size; (void)ws_size;

  const float* x  = (const float*)d_in[0];
  const int*   ei = (const int*)d_in[1];
  const float* enc_w1 = (const float*)d_in[2];
  const float* enc_b1 = (const float*)d_in[3];
  const float* enc_w2 = (const float*)d_in[4];
  const float* enc_b2 = (const float*)d_in[5];
  const float* env_g1_lin  = (const float*)d_in[6];
  const float* env_g1_asrc = (const float*)d_in[7];
  const float* env_g1_adst = (const float*)d_in[8];
  const float* env_g1_b    = (const float*)d_in[9];
  const float* env_g2_lin  = (const float*)d_in[10];
  const float* env_g2_asrc = (const float*)d_in[11];
  const float* env_g2_adst = (const float*)d_in[12];
  const float* env_g2_b    = (const float*)d_in[13];
  const float* bhv_g1_lin  = (const float*)d_in[14];
  const float* bhv_g1_asrc = (const float*)d_in[15];
  const float* bhv_g1_adst = (const float*)d_in[16];
  const float* bhv_g1_b    = (const float*)d_in[17];
  const float* bhv_g2_lin  = (const float*)d_in[18];
  const float* bhv_g2_asrc = (const float*)d_in[19];
  const float* bhv_g2_adst = (const float*)d_in[20];
  const float* bhv_g2_b    = (const float*)d_in[21];
  const float* phys_W0     = (const float*)d_in[22];
  const float* rnn_wih     = (const float*)d_in[23];
  const float* rnn_bih     = (const float*)d_in[24];
  const float* rnn_bhh     = (const float*)d_in[25];
  const float* phys_proj_w = (const float*)d_in[26];
  const float* phys_proj_b = (const float*)d_in[27];
  const float* gate_w1 = (const float*)d_in[28];
  const float* gate_b1 = (const float*)d_in[29];
  const float* gate_w2 = (const float*)d_in[30];
  const float* gate_b2 = (const float*)d_in[31];
  float* out = (float*)d_out;

  // workspace layout (fp32, 16B aligned slices)
  float* ws = (float*)d_ws;
  size_t off = 0;
  auto alloc = [&](size_t n) { float* p = ws + off; off += (n + 3) & ~(size_t)3; return p; };
  const size_t N = NNODES;
  float* xpad    = alloc(8 * N);        // x padded [N,8]
  float* ctx     = alloc(8 * N);
  float* tenc    = alloc(16 * N);
  float* xenv    = alloc(12 * N);       // [N,11] padded to stride 12
  float* xphys   = alloc(12 * N);       // [N,10] padded to stride 12
  float* xbehav  = alloc(12 * N);
  float* g1      = alloc(32 * N);
  float* gsc     = alloc(3 * N);
  float* envout  = alloc(16 * N);
  float* bhvout  = alloc(16 * N);
  float* physout = alloc(16 * N);
  float* hbuf    = alloc(32 * N);
  float* o1      = alloc(32 * N);
  float* aS      = alloc(2 * N);
  float* aD      = alloc(2 * N);
  unsigned* mbuf = (unsigned*)alloc(2 * N);
  float* zbuf    = alloc(2 * N);
  float* evS     = alloc(2 * N);
  float* xp      = alloc(10 * N);
  float* agg     = alloc(12 * N);       // [N,10] padded to stride 12
  float* BTp     = alloc(12 * 16);
  float* Bp      = alloc(32 * 32);      // weight staging (reused, stream-ordered)
  float* biasp   = alloc(32);
  float* zbias   = alloc(32);
  float* evE     = alloc(2 * (size_t)NEDGES);

  const int nb = (NNODES + 255) / 256;
  const int eb = (NEDGES + 255) / 256;
  const int tpb = 32 * WAVES_PER_BLK;

  hipMemsetAsync(zbias, 0, 32 * sizeof(float), stream);
  pad_x<<<nb, 256, 0, stream>>>(x, xpad);

  // shared encoder: ctx = relu(x@W1+b1)@W2+b2
  pad_wb<<<4, 256, 0, stream>>>(enc_w1, enc_b1, 6, 16, Bp, biasp, 8, 16);
  gemm_wmma_f32<8><<<gemm_grid(1), tpb, 0, stream>>>(xpad, 8, Bp, biasp,
                                                     tenc, 16, 16, /*relu*/1);
  pad_wb<<<4, 256, 0, stream>>>(enc_w2, enc_b2, 16, 8, Bp, biasp, 16, 16);
  gemm_wmma_f32<16><<<gemm_grid(1), tpb, 0, stream>>>(tenc, 16, Bp, biasp,
                                                      ctx, 8, 8, 0);
  prep_inputs<<<nb, 256, 0, stream>>>(x, ctx, xenv, xphys, xbehav);

  // gating scores (softmax fused into combine)
  pad_wb<<<4, 256, 0, stream>>>(gate_w1, gate_b1, 8, 32, Bp, biasp, 8, 32);
  gemm_wmma_f32<8><<<gemm_grid(2), tpb, 0, stream>>>(ctx, 8, Bp, biasp,
                                                     g1, 32, 32, /*relu*/1);
  pad_wb<<<4, 256, 0, stream>>>(gate_w2, gate_b2, 32, 3, Bp, biasp, 32, 16);
  gemm_wmma_f32<32><<<gemm_grid(1), tpb, 0, stream>>>(g1, 32, Bp, biasp,
                                                      gsc, 3, 3, 0);

  // env expert: GAT(11->32, H=2, elu) -> GAT(32->16, H=1)
  run_gat_layer(stream, ei, xenv, 12, 12, 2, 16, env_g1_lin, 11, env_g1_asrc,
                env_g1_adst, env_g1_b, Bp, biasp, hbuf, aS, aD, mbuf, zbuf,
                evE, evS, o1, /*elu*/2);
  run_gat_layer(stream, ei, o1, 32, 32, 1, 16, env_g2_lin, 32, env_g2_asrc,
                env_g2_adst, env_g2_b, Bp, biasp, hbuf, aS, aD, mbuf, zbuf,
                evE, evS, envout, 0);

  // behav expert
  run_gat_layer(stream, ei, xbehav, 12, 12, 2, 16, bhv_g1_lin, 11, bhv_g1_asrc,
                bhv_g1_adst, bhv_g1_b, Bp, biasp, hbuf, aS, aD, mbuf, zbuf,
                evE, evS, o1, /*elu*/2);
  run_gat_layer(stream, ei, o1, 32, 32, 1, 16, bhv_g2_lin, 32, bhv_g2_asrc,
                bhv_g2_adst, bhv_g2_b, Bp, biasp, hbuf, aS, aD, mbuf, zbuf,
                evE, evS, bhvout, 0);

  // phys expert: EvolveGCN-O step -> xp = x_phys@W_new^T -> segsum -> relu -> proj
  phys_rnn<<<1, 256, 0, stream>>>(phys_W0, rnn_wih, rnn_bih, rnn_bhh, BTp);
  gemm_wmma_f32<12><<<gemm_grid(1), tpb, 0, stream>>>(xphys, 12, BTp, zbias,
                                                      xp, 10, 10, 0);
  hipMemsetAsync(agg, 0, 12 * N * sizeof(float), stream);
  gcn_edge<<<eb, 256, 0, stream>>>(ei, xp, agg);
  relu_inplace<<<(unsigned)((12 * N + 255) / 256), 256, 0, stream>>>(agg, 12 * N);
  pad_wb<<<4, 256, 0, stream>>>(phys_proj_w, phys_proj_b, 10, 16, Bp, biasp, 12, 16);
  gemm_wmma_f32<12><<<gemm_grid(1), tpb, 0, stream>>>(agg, 12, Bp, biasp,
                                                      physout, 16, 16, 0);

  // softmax gate + mixture
  gate_combine<<<nb, 256, 0, stream>>>(gsc, envout, physout, bhvout, out);
}